// ShiftNMF_23364622090321
// MI455X (gfx1250) — compile-verified
//
#include <hip/hip_runtime.h>
#include <hip/hip_bf16.h>
#include <math.h>

#define PI_F 3.14159265358979323846f

typedef float v2f __attribute__((ext_vector_type(2)));
typedef float v8f __attribute__((ext_vector_type(8)));

__device__ __forceinline__ float softplus_f(float x) {
  return (x > 20.0f) ? x : log1pf(expf(x));
}

// A-operand tile of the 64-point DFT matrix from a 64-entry cos table:
//   C[m][k] = cos(2*pi*m*k/64) = cs64[(m*k)    & 63]
//   S[m][k] = sin(2*pi*m*k/64) = cs64[(m*k+48) & 63]   (sin x = cos(x - pi/2))
// 32-bit A 16x4 layout: lanes 0-15 hold K=0,1 in v0,v1 ; lanes 16-31 hold K=2,3.
__device__ __forceinline__ v2f dft_a_tile(const float* cs64, int ti, int kk,
                                          int lane, int sin_off) {
  int row = ti * 16 + (lane & 15);
  int k0  = kk * 4 + ((lane < 16) ? 0 : 2);
  v2f a;
  a.x = cs64[(row * k0 + sin_off) & 63];
  a.y = cs64[(row * (k0 + 1) + sin_off) & 63];
  return a;
}

// B-operand tile (4x16): element B(k, col) at Bbuf[k*rs + col*cs].
__device__ __forceinline__ v2f ld_b_tile(const float* B, int tj, int kk, int lane,
                                         int rs, int cs) {
  int col = tj * 16 + (lane & 15);
  int k0  = kk * 4 + ((lane < 16) ? 0 : 1);
  v2f b;
  b.x = B[(k0)     * rs + col * cs];
  b.y = B[(k0 + 2) * rs + col * cs];
  return b;
}

// acc += F64_tile(ti) * B_tile(tj)  over full K=64 (16 chained k-steps of 4)
__device__ __forceinline__ v8f dft_mm(const float* cs64, int sin_off, const float* B,
                                      int ti, int tj, int lane, int rs, int cs,
                                      v8f acc) {
#pragma unroll
  for (int kk = 0; kk < 16; ++kk) {
    v2f a = dft_a_tile(cs64, ti, kk, lane, sin_off);
    v2f b = ld_b_tile(B, tj, kk, lane, rs, cs);
    acc = __builtin_amdgcn_wmma_f32_16x16x4_f32(false, a, false, b,
                                                (short)0, acc, false, false);
  }
  return acc;
}

// ---------------------------------------------------------------------------
// Kernel 1: Hft[d, :] = FFT_4096(softplus(H[d, :])), four-step 64x64 via WMMA.
// One block per rank row d. 256 threads = 8 waves. ~48 KB LDS.
// ---------------------------------------------------------------------------
__global__ __launch_bounds__(256)
void shiftnmf_dft_kernel(const float* __restrict__ H, float2* __restrict__ Hft) {
  __shared__ float xlds[64 * 64];
  __shared__ float Yre[64 * 64];
  __shared__ float Yim[64 * 64];
  __shared__ float cs64[64];

  const int d    = blockIdx.x;
  const int tid  = threadIdx.x;
  const int lane = tid & 31;
  const int wave = tid >> 5;

  // 64-entry cos table (exact fp32 twiddles for the 64-point DFT matrix).
  if (tid < 64) cs64[tid] = cosf((2.0f * PI_F / 64.0f) * (float)tid);

  // Load row + softplus. x[k1][k2] row-major == natural order (k = 64*k1 + k2).
#pragma unroll
  for (int i = 0; i < 16; ++i) {
    int k = tid + i * 256;
    xlds[k] = softplus_f(H[d * 4096 + k]);
  }
  __syncthreads();

  // Step 1: Y[m1][k2] = sum_k1 (C - iS)[m1][k1] * x[k1][k2]   (x real)
  // Yre = C*x ; Yim = -(S*x). 16 output tiles over 8 waves.
  for (int t = wave; t < 16; t += 8) {
    int ti = t >> 2, tj = t & 3;
    v8f are = {0.f, 0.f, 0.f, 0.f, 0.f, 0.f, 0.f, 0.f};
    v8f aim = {0.f, 0.f, 0.f, 0.f, 0.f, 0.f, 0.f, 0.f};
    are = dft_mm(cs64, 0,  xlds, ti, tj, lane, 64, 1, are);
    aim = dft_mm(cs64, 48, xlds, ti, tj, lane, 64, 1, aim);
#pragma unroll
    for (int v = 0; v < 8; ++v) {
      int row = ti * 16 + ((lane < 16) ? v : v + 8);
      int col = tj * 16 + (lane & 15);
      Yre[row * 64 + col] = are[v];
      Yim[row * 64 + col] = -aim[v];
    }
  }
  __syncthreads();

  // Step 2: twiddle  Y'[m1][k2] *= exp(-2*pi*i * m1*k2 / 4096)
#pragma unroll
  for (int i = 0; i < 16; ++i) {
    int idx = tid + i * 256;
    int m1 = idx >> 6, k2 = idx & 63;
    float s, c;
    sincosf((-2.0f * PI_F / 4096.0f) * (float)(m1 * k2), &s, &c);
    float re = Yre[idx], im = Yim[idx];
    Yre[idx] = re * c - im * s;
    Yim[idx] = re * s + im * c;
  }
  __syncthreads();

  // Step 3: X[m1 + 64*m2] = sum_k2 (C - iS)[m2][k2] * Y'[m1][k2]
  // O[m2][m1]: B = Y'^T -> rs=1, cs=64.
  for (int t = wave; t < 16; t += 8) {
    int ti = t >> 2, tj = t & 3;  // ti: m2-tile, tj: m1-tile
    v8f o_re = {0.f, 0.f, 0.f, 0.f, 0.f, 0.f, 0.f, 0.f};
    o_re = dft_mm(cs64, 0,  Yre, ti, tj, lane, 1, 64, o_re);
    o_re = dft_mm(cs64, 48, Yim, ti, tj, lane, 1, 64, o_re);
    v8f o_ca = {0.f, 0.f, 0.f, 0.f, 0.f, 0.f, 0.f, 0.f};
    o_ca = dft_mm(cs64, 0,  Yim, ti, tj, lane, 1, 64, o_ca);
    v8f o_sb = {0.f, 0.f, 0.f, 0.f, 0.f, 0.f, 0.f, 0.f};
    o_sb = dft_mm(cs64, 48, Yre, ti, tj, lane, 1, 64, o_sb);
#pragma unroll
    for (int v = 0; v < 8; ++v) {
      int m2 = ti * 16 + ((lane < 16) ? v : v + 8);
      int m1 = tj * 16 + (lane & 15);
      float2 val;
      val.x = o_re[v];
      val.y = o_ca[v] - o_sb[v];
      Hft[d * 4096 + (m1 + (m2 << 6))] = val;
    }
  }
}

// ---------------------------------------------------------------------------
// Kernel 2: V[n,m] = sum_d softplus(W)[n,d] * exp(-2*pi*i*tau[n,d]*m/M) * Hft[d,m]
// One block per output row n; each thread owns 16 consecutive m.
// Phase factors advanced by complex rotation recurrence (2 sincos per (t,d)).
// ---------------------------------------------------------------------------
__global__ __launch_bounds__(256)
void shiftnmf_main_kernel(const float* __restrict__ W, const float* __restrict__ tau,
                          const float2* __restrict__ Hft, float2* __restrict__ out) {
  __shared__ float sw[32], tv[32], str[32], sti[32];
  const int n   = blockIdx.x;
  const int tid = threadIdx.x;

  if (tid < 32) {
    sw[tid] = softplus_f(W[n * 32 + tid]);
    float t = tau[n * 32 + tid];
    tv[tid] = t;
    float s, c;
    sincosf(-2.0f * PI_F * t / 4096.0f, &s, &c);
    str[tid] = c;
    sti[tid] = s;
  }
  __syncthreads();

  const int m0 = tid * 16;
  float2 acc[16];
#pragma unroll
  for (int j = 0; j < 16; ++j) { acc[j].x = 0.f; acc[j].y = 0.f; }

  for (int d = 0; d < 32; ++d) {
    const float swd = sw[d];
    const float t   = tv[d];
    float wr, wi;
    {
      float s0, c0;
      sincosf(-2.0f * PI_F * t * (float)m0 / 4096.0f, &s0, &c0);
      wr = c0; wi = s0;
    }
    const float sr = str[d], si = sti[d];
    const float2* hp = Hft + (d * 4096 + m0);
    if (d + 1 < 32) __builtin_prefetch((const void*)(hp + 4096), 0, 1);
#pragma unroll
    for (int j = 0; j < 16; ++j) {
      float2 h = hp[j];
      acc[j].x += swd * (wr * h.x - wi * h.y);
      acc[j].y += swd * (wr * h.y + wi * h.x);
      float nr = wr * sr - wi * si;
      float ni = wr * si + wi * sr;
      wr = nr; wi = ni;
    }
  }

  float2* op = out + (n * 4096 + m0);
#pragma unroll
  for (int j = 0; j < 16; ++j) op[j] = acc[j];
}

extern "C" void kernel_launch(void* const* d_in, const int* in_sizes, int n_in,
                              void* d_out, int out_size, void* d_ws, size_t ws_size,
                              hipStream_t stream) {
  (void)in_sizes; (void)n_in; (void)out_size; (void)ws_size;
  const float* W   = (const float*)d_in[0];  // (512, 32)
  const float* H   = (const float*)d_in[1];  // (32, 4096)
  const float* tau = (const float*)d_in[2];  // (512, 32)

  float2* Hft = (float2*)d_ws;               // 32 x 4096 complex = 1 MB scratch

  shiftnmf_dft_kernel<<<32, 256, 0, stream>>>(H, Hft);
  shiftnmf_main_kernel<<<512, 256, 0, stream>>>(W, tau, Hft, (float2*)d_out);
}